// PCALayer_41626823033071
// MI455X (gfx1250) — compile-verified
//
#include <hip/hip_runtime.h>

typedef __attribute__((ext_vector_type(2))) float v2f;
typedef __attribute__((ext_vector_type(8))) float v8f;

#define BATCH   8
#define NVOX    32768
#define CH      128
#define POUT    32
#define KSPLIT  16
#define ROWS_PER_BLOCK (NVOX / KSPLIT)   // 2048
#define CHUNK   64
#define TSTRIDE 136                       // 128 + 8 pad -> lanes16-31 hit disjoint banks

// ---------------------------------------------------------------------------
// Kernel 1: partial Gram G_part = X_chunk^T X_chunk  (per batch, K-split) +
// partial column sums. Uses V_WMMA_F32_16X16X4_F32 (fp32 matrix pipe).
// ---------------------------------------------------------------------------
__global__ __launch_bounds__(256) void gram_kernel(const float* __restrict__ x,
                                                   float* __restrict__ Gpart,
                                                   float* __restrict__ Spart) {
    __shared__ float tile[CHUNK * TSTRIDE];
    const int b    = blockIdx.x / KSPLIT;
    const int kb   = blockIdx.x % KSPLIT;
    const int tid  = threadIdx.x;
    const int lane = tid & 31;
    const int w    = tid >> 5;                       // wave 0..7 -> output tile-row
    const float* xb = x + ((size_t)b * NVOX + (size_t)kb * ROWS_PER_BLOCK) * CH;

    v8f acc[8];
#pragma unroll
    for (int n = 0; n < 8; ++n) acc[n] = {};
    float4 csum = make_float4(0.f, 0.f, 0.f, 0.f);
    const int cgrp = (tid & 31) * 4;                 // fixed channel group per thread

    const int m    = lane & 15;
    const int ksel = (lane >> 4) * 2;
    const int acol = w * 16 + m;                     // A-fragment channel (per-wave, no reg indexing)

    for (int chunk = 0; chunk < ROWS_PER_BLOCK / CHUNK; ++chunk) {
        const float4* src = (const float4*)(xb + (size_t)chunk * CHUNK * CH);
        __syncthreads();
#pragma unroll
        for (int i = 0; i < 8; ++i) {
            int e   = tid + i * 256;                 // float4 index in 64x128 chunk
            int row = e >> 5;
            float4 v = src[e];
            *((float4*)&tile[row * TSTRIDE + cgrp]) = v;
            csum.x += v.x; csum.y += v.y; csum.z += v.z; csum.w += v.w;
        }
        __syncthreads();
#pragma unroll
        for (int kk = 0; kk < CHUNK / 4; ++kk) {
            int krow = kk * 4 + ksel;                // K: lanes0-15 {k,k+1}, lanes16-31 {k+2,k+3}
            v2f a;                                   // direct LDS load -> no cndmask select tree
            a.x = tile[(krow + 0) * TSTRIDE + acol];
            a.y = tile[(krow + 1) * TSTRIDE + acol];
            v2f frag[8];
#pragma unroll
            for (int n = 0; n < 8; ++n) {            // B fragments for all 8 channel groups
                frag[n].x = tile[(krow + 0) * TSTRIDE + n * 16 + m];
                frag[n].y = tile[(krow + 1) * TSTRIDE + n * 16 + m];
            }
#pragma unroll
            for (int n = 0; n < 8; ++n)
                acc[n] = __builtin_amdgcn_wmma_f32_16x16x4_f32(
                    false, a, false, frag[n], (short)0, acc[n], false, false);
        }
    }

    // Plain (deterministic) partial store
    float* Gb = Gpart + ((size_t)(kb * BATCH + b)) * CH * CH;
#pragma unroll
    for (int n = 0; n < 8; ++n) {
#pragma unroll
        for (int r = 0; r < 8; ++r) {
            int row = w * 16 + ((lane < 16) ? r : (8 + r));
            int col = n * 16 + m;
            Gb[row * CH + col] = acc[n][r];
        }
    }

    // block-level reduction of column sums (deterministic order)
    __syncthreads();
    ((float4*)tile)[tid] = csum;
    __syncthreads();
    if (tid < CH) {
        int g = tid >> 2, comp = tid & 3;
        float s = 0.f;
        for (int i = 0; i < 8; ++i) s += tile[(g + 32 * i) * 4 + comp];
        Spart[(kb * BATCH + b) * CH + tid] = s;
    }
}

// ---------------------------------------------------------------------------
// Kernel 2: cov = sum_k Gpart - s s^T / N   (fixed-order reduction)
// ---------------------------------------------------------------------------
__global__ __launch_bounds__(256) void cov_finalize(const float* __restrict__ Gpart,
                                                    const float* __restrict__ Spart,
                                                    float* __restrict__ cov,
                                                    float* __restrict__ Ssum) {
    __shared__ float sloc[CH];
    const int b = blockIdx.x, tid = threadIdx.x;
    if (tid < CH) {
        float s = 0.f;
        for (int k = 0; k < KSPLIT; ++k) s += Spart[(k * BATCH + b) * CH + tid];
        sloc[tid] = s;
        Ssum[b * CH + tid] = s;
    }
    __syncthreads();
    const float inv = 1.0f / (float)NVOX;
    for (int idx = tid; idx < CH * CH; idx += 256) {
        float g = 0.f;
        for (int k = 0; k < KSPLIT; ++k)
            g += Gpart[((size_t)(k * BATCH + b)) * CH * CH + idx];
        int i = idx >> 7, j = idx & 127;
        cov[b * CH * CH + idx] = g - sloc[i] * sloc[j] * inv;
    }
}

// ---------------------------------------------------------------------------
// Kernel 3: parallel cyclic Jacobi eigensolver, one workgroup per batch.
// A,V in dynamic LDS (2 * 128 * 129 * 4B = 132 KB; CDNA5 allows 320 KB/WGP).
// ---------------------------------------------------------------------------
#define JTHREADS 512
#define NSWEEP   10
#define ASTR     129

__global__ __launch_bounds__(JTHREADS) void jacobi_kernel(const float* __restrict__ covB,
                                                          const float* __restrict__ Ssum,
                                                          float* __restrict__ Vtop,
                                                          float* __restrict__ Tproj) {
    extern __shared__ float smem[];
    float* A = smem;                  // [128][129]
    float* V = smem + CH * ASTR;      // [128][129]
    __shared__ float cs_[64], sn_[64];
    __shared__ int   pr_[64], qr_[64];
    __shared__ float diag_[CH];
    __shared__ int   rank_[CH];

    const int b = blockIdx.x, tid = threadIdx.x;
    const float* cov = covB + (size_t)b * CH * CH;

    for (int idx = tid; idx < CH * CH; idx += JTHREADS) {
        int i = idx >> 7, j = idx & 127;
        A[i * ASTR + j] = cov[idx];
        V[i * ASTR + j] = (i == j) ? 1.0f : 0.0f;
    }
    __syncthreads();

    for (int sweep = 0; sweep < NSWEEP; ++sweep) {
        for (int r = 0; r < CH - 1; ++r) {
            if (tid < 64) {                          // round-robin pairing + rotation
                int k = tid, p, q;
                if (k == 0) { p = r % (CH - 1); q = CH - 1; }
                else { p = (r + k) % (CH - 1); q = (r + (CH - 1) - k) % (CH - 1); }
                if (p > q) { int t0 = p; p = q; q = t0; }
                float app = A[p * ASTR + p], aqq = A[q * ASTR + q], apq = A[p * ASTR + q];
                float c = 1.0f, s = 0.0f;
                if (fabsf(apq) > 1e-12f * (fabsf(app) + fabsf(aqq)) + 1e-30f) {
                    float tau = (aqq - app) / (2.0f * apq);
                    float t   = copysignf(1.0f, tau) / (fabsf(tau) + sqrtf(1.0f + tau * tau));
                    c = 1.0f / sqrtf(1.0f + t * t);
                    s = t * c;
                }
                pr_[k] = p; qr_[k] = q; cs_[k] = c; sn_[k] = s;
            }
            __syncthreads();
            {   // rows: A <- J^T A      (thread = column j, pair-slice h)
                int j = tid & 127, h = tid >> 7;     // h: 0..3
                for (int k = h * 16; k < h * 16 + 16; ++k) {
                    int p = pr_[k], q = qr_[k];
                    float c = cs_[k], s = sn_[k];
                    float a0 = A[p * ASTR + j], a1 = A[q * ASTR + j];
                    A[p * ASTR + j] = c * a0 - s * a1;
                    A[q * ASTR + j] = s * a0 + c * a1;
                }
            }
            __syncthreads();
            {   // cols: A <- A J, V <- V J   (thread = pair k, row-slice h)
                int k = tid & 63, h = tid >> 6;      // h: 0..7
                int p = pr_[k], q = qr_[k];
                float c = cs_[k], s = sn_[k];
                for (int i = h * 16; i < h * 16 + 16; ++i) {
                    float a0 = A[i * ASTR + p], a1 = A[i * ASTR + q];
                    A[i * ASTR + p] = c * a0 - s * a1;
                    A[i * ASTR + q] = s * a0 + c * a1;
                    float v0 = V[i * ASTR + p], v1 = V[i * ASTR + q];
                    V[i * ASTR + p] = c * v0 - s * v1;
                    V[i * ASTR + q] = s * v0 + c * v1;
                }
            }
            __syncthreads();
        }
    }

    if (tid < CH) diag_[tid] = A[tid * ASTR + tid];
    __syncthreads();
    if (tid < CH) {                                  // ascending rank (eigh order)
        float d = diag_[tid]; int rnk = 0;
        for (int j = 0; j < CH; ++j) {
            float dj = diag_[j];
            if (dj < d || (dj == d && j < tid)) rnk++;
        }
        rank_[tid] = rnk;
    }
    __syncthreads();
    if (tid < CH) {                                  // top-32 columns, ascending
        int rnk = rank_[tid];
        if (rnk >= CH - POUT) {
            int oc = rnk - (CH - POUT);
            for (int i = 0; i < CH; ++i)
                Vtop[((size_t)b * CH + i) * POUT + oc] = V[i * ASTR + tid];
        }
    }
    __syncthreads();
    if (tid < POUT) {                                // t = mu^T Vtop
        int target = CH - POUT + tid, col = 0;
        for (int j = 0; j < CH; ++j) if (rank_[j] == target) col = j;
        float acc = 0.f;
        const float invN = 1.0f / (float)NVOX;
        for (int c = 0; c < CH; ++c)
            acc += Ssum[b * CH + c] * invN * V[c * ASTR + col];
        Tproj[b * POUT + tid] = acc;
    }
}

// ---------------------------------------------------------------------------
// Kernel 4: out = X * Vtop - t   via WMMA f32 16x16x4
// ---------------------------------------------------------------------------
#define PROWS 64
#define XSTR  132     // 4m mod 64 distinct -> conflict-free A-fragment reads
#define VSTR  48      // +2 rows => +96 => bank shift 32 -> disjoint halves

__global__ __launch_bounds__(128) void proj_kernel(const float* __restrict__ x,
                                                   const float* __restrict__ Vtop,
                                                   const float* __restrict__ Tproj,
                                                   float* __restrict__ out) {
    __shared__ float xt[PROWS * XSTR];
    __shared__ float vl[CH * VSTR];
    __shared__ float tl[POUT];
    const int nb   = NVOX / PROWS;                   // 512 row-blocks per batch
    const int b    = blockIdx.x / nb;
    const int rb   = blockIdx.x % nb;
    const int tid  = threadIdx.x;
    const int lane = tid & 31;
    const int w    = tid >> 5;                       // wave 0..3 -> 16-row strip
    const float* xb = x + ((size_t)b * NVOX + (size_t)rb * PROWS) * CH;

    const float4* v4 = (const float4*)(Vtop + (size_t)b * CH * POUT);
#pragma unroll
    for (int i = 0; i < 8; ++i) {                    // 128x32 V -> LDS (padded)
        int e = tid + i * 128;
        int krow = e >> 3, cg = (e & 7) * 4;
        *((float4*)&vl[krow * VSTR + cg]) = v4[e];
    }
    if (tid < POUT) tl[tid] = Tproj[b * POUT + tid];
    const float4* x4 = (const float4*)xb;
#pragma unroll
    for (int i = 0; i < 16; ++i) {                   // 64x128 X tile -> LDS
        int e = tid + i * 128;
        int row = e >> 5, cg = (e & 31) * 4;
        *((float4*)&xt[row * XSTR + cg]) = x4[e];
    }
    __syncthreads();

    v8f acc0 = {}, acc1 = {};
    const int m    = lane & 15;
    const int ksel = (lane >> 4) * 2;
    const int rowb = w * 16;
#pragma unroll
    for (int kk = 0; kk < CH / 4; ++kk) {
        int k = kk * 4 + ksel;
        v2f a, b0, b1;
        a.x  = xt[(rowb + m) * XSTR + k + 0];
        a.y  = xt[(rowb + m) * XSTR + k + 1];
        b0.x = vl[(k + 0) * VSTR + m];
        b0.y = vl[(k + 1) * VSTR + m];
        b1.x = vl[(k + 0) * VSTR + 16 + m];
        b1.y = vl[(k + 1) * VSTR + 16 + m];
        acc0 = __builtin_amdgcn_wmma_f32_16x16x4_f32(false, a, false, b0, (short)0, acc0, false, false);
        acc1 = __builtin_amdgcn_wmma_f32_16x16x4_f32(false, a, false, b1, (short)0, acc1, false, false);
    }

    float* ob = out + ((size_t)b * NVOX + (size_t)rb * PROWS) * POUT;
#pragma unroll
    for (int r = 0; r < 8; ++r) {
        int row = rowb + ((lane < 16) ? r : 8 + r);
        ob[row * POUT + m]      = acc0[r] - tl[m];
        ob[row * POUT + 16 + m] = acc1[r] - tl[16 + m];
    }
}

// ---------------------------------------------------------------------------
extern "C" void kernel_launch(void* const* d_in, const int* in_sizes, int n_in,
                              void* d_out, int out_size, void* d_ws, size_t ws_size,
                              hipStream_t stream) {
    (void)in_sizes; (void)n_in; (void)out_size; (void)ws_size;
    const float* x = (const float*)d_in[0];
    float* out = (float*)d_out;

    float* Gpart = (float*)d_ws;                                   // 16*8*16384
    float* Spart = Gpart + (size_t)KSPLIT * BATCH * CH * CH;       // 16*8*128
    float* cov   = Spart + (size_t)KSPLIT * BATCH * CH;            // 8*16384
    float* Ssum  = cov   + (size_t)BATCH * CH * CH;                // 8*128
    float* Vt    = Ssum  + (size_t)BATCH * CH;                     // 8*128*32
    float* Tp    = Vt    + (size_t)BATCH * CH * POUT;              // 8*32

    gram_kernel<<<BATCH * KSPLIT, 256, 0, stream>>>(x, Gpart, Spart);
    cov_finalize<<<BATCH, 256, 0, stream>>>(Gpart, Spart, cov, Ssum);
    jacobi_kernel<<<BATCH, JTHREADS, 2 * CH * ASTR * sizeof(float), stream>>>(cov, Ssum, Vt, Tp);
    proj_kernel<<<BATCH * (NVOX / PROWS), 128, 0, stream>>>(x, Vt, Tp, out);
}